// NeRFAggregator_60404420051272
// MI455X (gfx1250) — compile-verified
//
#include <hip/hip_runtime.h>
#include <stdint.h>

#define NRAYS   (8 * 8192)      // B*R
#define NP      128             // samples per ray
#define FARV    1e10f
#define EPSV    1e-10f
#define WPB     8               // waves per block (256 threads)

// ---- CDNA5 async global->LDS copy (ASYNCcnt-tracked DMA path) ----
__device__ __forceinline__ void async_load_b128(uint32_t lds_off, const void* gaddr) {
    asm volatile("global_load_async_to_lds_b128 %0, %1, off"
                 :: "v"(lds_off), "v"(gaddr)
                 : "memory");
}

template<int N>
__device__ __forceinline__ void wait_async() {
    asm volatile("s_wait_asynccnt %0" :: "i"(N) : "memory");
}

__device__ __forceinline__ float fast_sigmoid(float x) {
    // v_exp_f32 + v_rcp_f32 (2 trans ops) instead of libm div sequence
    float e = __expf(-x);
    return __builtin_amdgcn_rcpf(1.0f + e);
}

// One chunk = 32 consecutive samples of one ray, lane == sample-in-chunk.
#define CHUNK(I, WAITN, ZNEXT)                                                  \
    do {                                                                        \
        wait_async<WAITN>();  /* chunk I's async DMA has landed in LDS */       \
        const float4 rv = *(const float4*)(smem + lbase + (I)*512 + lane*16);   \
        float zc = zv[I];                                                       \
        float zn = __shfl_down(zc, 1, 32);                                      \
        float znc = __shfl((ZNEXT), 0, 32);                                     \
        if (lane == 31) zn = znc;                                               \
        const int s = (I)*32 + lane;                                            \
        float dd = (s == NP - 1) ? FARV : (zn - zc);                            \
        dd *= nrm;                                                              \
        float sg    = fmaxf(rv.w, 0.0f);                                        \
        float alpha = 1.0f - __expf(-sg * dd);                                  \
        /* wave32 Hillis-Steele inclusive prefix product of (1-alpha+eps) */    \
        float incl = 1.0f - alpha + EPSV;                                       \
        _Pragma("unroll")                                                       \
        for (int off = 1; off < 32; off <<= 1) {                                \
            float v = __shfl_up(incl, off, 32);                                 \
            if (lane >= off) incl *= v;                                         \
        }                                                                       \
        float excl = __shfl_up(incl, 1, 32);                                    \
        if (lane == 0) excl = 1.0f;                                             \
        float trans = carry * excl;                                             \
        float w     = alpha * trans;                                            \
        carry *= __shfl(incl, 31, 32);                                          \
        acc_r += w * fast_sigmoid(rv.x);                                        \
        acc_g += w * fast_sigmoid(rv.y);                                        \
        acc_b += w * fast_sigmoid(rv.z);                                        \
        acc_w += w;                                                             \
        acc_z += w * zc;                                                        \
        __builtin_nontemporal_store(w, out_w + (size_t)ray * NP + s);           \
    } while (0)

__global__ __launch_bounds__(256) void nerf_agg_kernel(
    const float* __restrict__ raw,     // (B,R,P,4)
    const float* __restrict__ z,       // (B,R,P)
    const float* __restrict__ rays_d,  // (B,R,3)
    const float* __restrict__ bg,      // (3,)
    float* __restrict__ out)
{
    // 8 waves * 4 chunks * 512B = 16 KB staging (LDS is 320 KB/WGP on CDNA5)
    __shared__ __align__(16) unsigned char smem[WPB * NP * 16];

    const int lane = threadIdx.x & 31;
    const int wave = threadIdx.x >> 5;
    const int ray  = blockIdx.x * WPB + wave;

    const float*   ray_raw = raw + (size_t)ray * (NP * 4);
    const uint32_t lbase   = (uint32_t)wave * 2048u;

    // Kick off the whole ray's raw stream as 4 async DMA chunks (512B each,
    // 32 lanes x b128, perfectly coalesced). ASYNCcnt completes in order.
#pragma unroll
    for (int i = 0; i < 4; ++i) {
        async_load_b128(lbase + (uint32_t)i * 512u + (uint32_t)lane * 16u,
                        (const char*)ray_raw + i * 512 + lane * 16);
    }

    // z: 4 coalesced non-temporal b32 loads (read-once stream)
    const float* zr = z + (size_t)ray * NP;
    float zv[4];
#pragma unroll
    for (int i = 0; i < 4; ++i)
        zv[i] = __builtin_nontemporal_load(zr + i * 32 + lane);

    // |rays_d| (uniform per wave; tiny)
    const float dx = rays_d[ray * 3 + 0];
    const float dy = rays_d[ray * 3 + 1];
    const float dz = rays_d[ray * 3 + 2];
    const float nrm = sqrtf(dx * dx + dy * dy + dz * dz);

    float* out_w = out + (size_t)6 * NRAYS;   // weights block

    float carry = 1.0f;  // running exclusive transmittance across chunks
    float acc_r = 0.f, acc_g = 0.f, acc_b = 0.f, acc_w = 0.f, acc_z = 0.f;

    CHUNK(0, 3, zv[1]);
    CHUNK(1, 2, zv[2]);
    CHUNK(2, 1, zv[3]);
    CHUNK(3, 0, zv[3]);   // lane31 of last chunk uses FAR; ZNEXT unused

    // 5-value wave32 butterfly reduction
#pragma unroll
    for (int off = 16; off > 0; off >>= 1) {
        acc_r += __shfl_xor(acc_r, off, 32);
        acc_g += __shfl_xor(acc_g, off, 32);
        acc_b += __shfl_xor(acc_b, off, 32);
        acc_w += __shfl_xor(acc_w, off, 32);
        acc_z += __shfl_xor(acc_z, off, 32);
    }

    if (lane == 0) {
        const float b0 = bg[0], b1 = bg[1], b2 = bg[2];
        const bool is_bg = (b0 >= 0.f && b0 <= 1.f) &&
                           (b1 >= 0.f && b1 <= 1.f) &&
                           (b2 >= 0.f && b2 <= 1.f);
        float r = acc_r, g = acc_g, b = acc_b;
        if (is_bg) {
            const float o = 1.0f - acc_w;
            r += o * b0; g += o * b1; b += o * b2;
        }
        const float ratio = acc_z / acc_w;
        const float disp  = 1.0f / (fmaxf(ratio - EPSV, 0.0f) + EPSV);

        out[(size_t)ray * 3 + 0] = r;                 // rgb_map
        out[(size_t)ray * 3 + 1] = g;
        out[(size_t)ray * 3 + 2] = b;
        out[(size_t)3 * NRAYS + ray] = acc_z;         // depth_map
        out[(size_t)4 * NRAYS + ray] = disp;          // disparity_map
        out[(size_t)5 * NRAYS + ray] = acc_w;         // alpha_map
    }
}

extern "C" void kernel_launch(void* const* d_in, const int* in_sizes, int n_in,
                              void* d_out, int out_size, void* d_ws, size_t ws_size,
                              hipStream_t stream) {
    const float* raw    = (const float*)d_in[0];
    const float* z      = (const float*)d_in[1];
    const float* rays_d = (const float*)d_in[2];
    const float* bg     = (const float*)d_in[3];
    float* out = (float*)d_out;

    dim3 grid(NRAYS / WPB);   // 8192 blocks, 256 threads = 8 wave32 per block
    nerf_agg_kernel<<<grid, 256, 0, stream>>>(raw, z, rays_d, bg, out);
}